// RoPE_9036611190976
// MI455X (gfx1250) — compile-verified
//
#include <hip/hip_runtime.h>
#include <stdint.h>

// ---------------------------------------------------------------------------
// RoPE on MI455X (gfx1250).
//
// out[b,h,s,2k]   = cos(s,k)*x[b,h,s,2k] - sin(s,k)*x[b,h,s,2k+1]
// out[b,h,s,2k+1] = sin(s,k)*x[b,h,s,2k] + cos(s,k)*x[b,h,s,2k+1]
// with cos(s,k) = R[pos[s], 2k, 2k], sin(s,k) = R[pos[s], 2k+1, 2k].
//
// Memory-bound: 256 MiB of mandatory traffic -> ~11us floor at 23.3 TB/s.
// We never touch the dense (mostly-zero) R table; the TDM gathers its
// diagonal (stride 1032 B) straight into LDS via tensor_load_to_lds.
// ---------------------------------------------------------------------------

typedef float        f32x4 __attribute__((ext_vector_type(4)));
typedef unsigned int u32x4 __attribute__((ext_vector_type(4)));
typedef int          i32x4 __attribute__((ext_vector_type(4)));
typedef int          i32x8 __attribute__((ext_vector_type(8)));

#define DK        128
#define SEQ       4096
#define NROWS     64                          // 4 batches * 16 heads
#define ROWSTRIDE ((size_t)SEQ * DK)          // (b,h)->(b,h+1) stride in floats

// Issue one TDM descriptor: walk 64 elements with a 258-element (1032 B)
// stride starting at gaddr, depositing them contiguously at LDS byte ldsOff.
// 2D tile: tile_dim0=1 (x), tile_dim1=64 (y), tensor_dim0_stride=258.
__device__ __forceinline__ void tdm_gather_diag64(unsigned ldsOff, uint64_t gaddr) {
    u32x4 g0;
    g0[0] = 1u;                                        // count=1 (valid), user mode
    g0[1] = ldsOff;                                    // LDS byte address of tile
    g0[2] = (unsigned)(gaddr & 0xFFFFFFFFu);           // global_addr[31:0]
    g0[3] = (unsigned)((gaddr >> 32) & 0x1FFFFFFu)     // global_addr[56:32]
          | (2u << 30);                                // type = 2 ("image")

    i32x8 g1;
    g1[0] = (2 << 16);      // data_size=2 -> 4 bytes; workgroup_mask=0; no pad/iter
    g1[1] = (1 << 16);      // tensor_dim0[15:0] = 1   (bits 63:48)
    g1[2] = (64 << 16);     // tensor_dim1[15:0] = 64  (bits 95:80)
    g1[3] = (1 << 16);      // tile_dim0 = 1           (bits 127:112)
    g1[4] = 64;             // tile_dim1 = 64; tile_dim2 = 0
    g1[5] = 258;            // tensor_dim0_stride[31:0] = 258 elems = 1032 B
    g1[6] = 0;              // stride0[47:32]=0, stride1[15:0]=0
    g1[7] = 0;

    i32x4 gz = {0, 0, 0, 0};                           // groups 2/3: dims unused
#if __clang_major__ >= 23
    i32x8 gz8 = {0, 0, 0, 0, 0, 0, 0, 0};
    __builtin_amdgcn_tensor_load_to_lds(g0, g1, gz, gz, gz8, 0);
#else
    __builtin_amdgcn_tensor_load_to_lds(g0, g1, gz, gz, 0);
#endif
}

__global__ void __launch_bounds__(256)
rope_apply(const float* __restrict__ x,
           const int*   __restrict__ tpos,
           const float* __restrict__ R,
           float*       __restrict__ out) {
    // tbl[0..63] = cos_k, tbl[64..127] = sin_k for this position
    __shared__ __align__(16) float tbl[2 * (DK / 2)];

    const int s = blockIdx.x;                          // sequence position

    if (threadIdx.x == 0) {
        const int pos = tpos[s];                       // uniform -> SGPR
        const uint64_t rbase =
            (uint64_t)(uintptr_t)(R + (size_t)pos * (DK * DK));
        const unsigned lds0 = (unsigned)(uintptr_t)(&tbl[0]);
        // cos: R[pos][2k][2k]   -> byte offset 1032*k
        tdm_gather_diag64(lds0, rbase);
        // sin: R[pos][2k+1][2k] -> byte offset 512 + 1032*k
        tdm_gather_diag64(lds0 + 256, rbase + 512);
        __builtin_amdgcn_s_wait_tensorcnt(0);          // both TDM ops done
    }
    __syncthreads();                                   // publish LDS to all waves

    const int lane = threadIdx.x & 31;
    const int wv   = threadIdx.x >> 5;                 // 8 waves per block

    // Lane covers d = 4*lane .. 4*lane+3  =>  pairs k0 = 2*lane, k1 = 2*lane+1
    const float c0 = tbl[2 * lane];
    const float c1 = tbl[2 * lane + 1];
    const float s0 = tbl[64 + 2 * lane];
    const float s1 = tbl[64 + 2 * lane + 1];

    const size_t colOff = (size_t)s * DK + (size_t)(lane * 4);

    // 64 (b,h) rows; wave w handles rows w, w+8, ..., fully unrolled so the
    // compiler keeps 8 b128 loads in flight per wave.
#pragma unroll
    for (int i = 0; i < NROWS / 8; ++i) {
        const int r = wv + i * 8;
        const size_t off = (size_t)r * ROWSTRIDE + colOff;

        f32x4 v = __builtin_nontemporal_load((const f32x4*)(x + off));
        f32x4 o;
        o[0] = c0 * v[0] - s0 * v[1];
        o[1] = s0 * v[0] + c0 * v[1];
        o[2] = c1 * v[2] - s1 * v[3];
        o[3] = s1 * v[2] + c1 * v[3];
        __builtin_nontemporal_store(o, (f32x4*)(out + off));
    }
}

extern "C" void kernel_launch(void* const* d_in, const int* in_sizes, int n_in,
                              void* d_out, int out_size, void* d_ws, size_t ws_size,
                              hipStream_t stream) {
    (void)in_sizes; (void)n_in; (void)out_size; (void)d_ws; (void)ws_size;
    const float* x    = (const float*)d_in[0];
    const int*   tpos = (const int*)  d_in[1];
    const float* R    = (const float*)d_in[2];
    float*       out  = (float*)d_out;

    rope_apply<<<dim3(SEQ), dim3(256), 0, stream>>>(x, tpos, R, out);
}